// GaussianRenderer_20452634264087
// MI455X (gfx1250) — compile-verified
//
#include <hip/hip_runtime.h>
#include <math.h>

#define NGAUSS 512
#define IMG_H 256
#define IMG_W 256
#define NPIX (IMG_H * IMG_W)

typedef __attribute__((ext_vector_type(2))) float v2f;
typedef __attribute__((ext_vector_type(4))) float v4f;
typedef __attribute__((ext_vector_type(8))) float v8f;
typedef __attribute__((ext_vector_type(4))) int   i4;

// Workspace layout (floats), contiguous so it can be staged to LDS in one sweep:
//   [0, 512*8)           : sorted quadratic coeffs, coef[i*8 + k], k=0..5 used, 6..7 zero
//   [4096, 4608)         : sorted mult = opacity * valid
//   [4608, 6144)         : sorted colors rgb
#define WS_COEF 0
#define WS_MULT (NGAUSS * 8)
#define WS_COLS (NGAUSS * 9)
#define WS_TOTAL (NGAUSS * 12)          // 6144 floats = 24 KB = 1536 x b128

#if defined(__AMDGCN__) && __has_builtin(__builtin_amdgcn_global_load_async_to_lds_b128) && \
    __has_builtin(__builtin_amdgcn_s_wait_asynccnt)
#define GS_USE_ASYNC_LDS 1
#endif

// ---------------------------------------------------------------------------
// Kernel 1: per-gaussian projection + 2D covariance -> quadratic coefficients,
// then bitonic sort by depth in LDS, write sorted tables to workspace.
// ---------------------------------------------------------------------------
__global__ __launch_bounds__(NGAUSS) void gs_prep_sort_kernel(
    const float* __restrict__ means3D, const float* __restrict__ covs3d,
    const float* __restrict__ colors, const float* __restrict__ opacities,
    const float* __restrict__ Km, const float* __restrict__ Rm,
    const float* __restrict__ tv, float* __restrict__ ws)
{
    __shared__ float skey[NGAUSS];
    __shared__ int   sidx[NGAUSS];
    __shared__ float sval[NGAUSS][12];   // c0..c5, mult, r, g, b

    const int n = threadIdx.x;

    // --- camera transform ---
    const float mx = means3D[n * 3 + 0], my = means3D[n * 3 + 1], mz = means3D[n * 3 + 2];
    const float R00 = Rm[0], R01 = Rm[1], R02 = Rm[2];
    const float R10 = Rm[3], R11 = Rm[4], R12 = Rm[5];
    const float R20 = Rm[6], R21 = Rm[7], R22 = Rm[8];
    const float cx = R00 * mx + R01 * my + R02 * mz + tv[0];
    const float cy = R10 * mx + R11 * my + R12 * mz + tv[1];
    const float cz = R20 * mx + R21 * my + R22 * mz + tv[2];
    const float depth = fmaxf(cz, 1.0f);

    // --- screen projection (cam @ K^T) ---
    const float s0 = Km[0] * cx + Km[1] * cy + Km[2] * cz;
    const float s1 = Km[3] * cx + Km[4] * cy + Km[5] * cz;
    const float s2 = Km[6] * cx + Km[7] * cy + Km[8] * cz;
    const float m2x = s0 / s2;
    const float m2y = s1 / s2;

    // --- Jacobian of projection (uses unclipped z, as in reference) ---
    const float fx = Km[0], fy = Km[4];
    const float iz = 1.0f / cz;
    const float J00 = fx * iz, J02 = -fx * cx * iz * iz;
    const float J11 = fy * iz, J12 = -fy * cy * iz * iz;

    // --- covs_cam = R * Sigma * R^T ---
    const float S00 = covs3d[n * 9 + 0], S01 = covs3d[n * 9 + 1], S02 = covs3d[n * 9 + 2];
    const float S10 = covs3d[n * 9 + 3], S11 = covs3d[n * 9 + 4], S12 = covs3d[n * 9 + 5];
    const float S20 = covs3d[n * 9 + 6], S21 = covs3d[n * 9 + 7], S22 = covs3d[n * 9 + 8];
    const float M00 = R00 * S00 + R01 * S10 + R02 * S20;
    const float M01 = R00 * S01 + R01 * S11 + R02 * S21;
    const float M02 = R00 * S02 + R01 * S12 + R02 * S22;
    const float M10 = R10 * S00 + R11 * S10 + R12 * S20;
    const float M11 = R10 * S01 + R11 * S11 + R12 * S21;
    const float M12 = R10 * S02 + R11 * S12 + R12 * S22;
    const float M20 = R20 * S00 + R21 * S10 + R22 * S20;
    const float M21 = R20 * S01 + R21 * S11 + R22 * S21;
    const float M22 = R20 * S02 + R21 * S12 + R22 * S22;
    const float V00 = M00 * R00 + M01 * R01 + M02 * R02;
    const float V01 = M00 * R10 + M01 * R11 + M02 * R12;
    const float V02 = M00 * R20 + M01 * R21 + M02 * R22;
    const float V10 = M10 * R00 + M11 * R01 + M12 * R02;
    const float V11 = M10 * R10 + M11 * R11 + M12 * R12;
    const float V12 = M10 * R20 + M11 * R21 + M12 * R22;
    const float V20 = M20 * R00 + M21 * R01 + M22 * R02;
    const float V21 = M20 * R10 + M21 * R11 + M22 * R12;
    const float V22 = M20 * R20 + M21 * R21 + M22 * R22;

    // --- covs2D = J * V * J^T,  J = [[J00,0,J02],[0,J11,J12]] ---
    const float t00 = J00 * V00 + J02 * V20;
    const float t01 = J00 * V01 + J02 * V21;
    const float t02 = J00 * V02 + J02 * V22;
    const float t10 = J11 * V10 + J12 * V20;
    const float t11 = J11 * V11 + J12 * V21;
    const float t12 = J11 * V12 + J12 * V22;
    const float C00 = t00 * J00 + t02 * J02;
    const float C01 = t01 * J11 + t02 * J12;
    const float C10 = t10 * J00 + t12 * J02;
    const float C11 = t11 * J11 + t12 * J12;

    const float eps = 1e-4f;
    const float a = C00 + eps, b = C01, c = C10, d = C11 + eps;
    const float det  = a * d - b * c;
    const float idet = 1.0f / det;
    const float i00 = d * idet, i01 = -b * idet, i10 = -c * idet, i11 = a * idet;
    const float sxy = i01 + i10;
    const float norm = 1.0f / (6.283185307179586f * sqrtf(det));

    const bool valid = (depth > 1.0f) && (depth < 50.0f);
    const float mult = valid ? opacities[n] : 0.0f;

    // P(u,v) = c0*u^2 + c1*u*v + c2*v^2 + c3*u + c4*v + c5   (ln(norm) folded in)
    const float c0 = -0.5f * i00;
    const float c1 = -0.5f * sxy;
    const float c2 = -0.5f * i11;
    const float c3 = i00 * m2x + 0.5f * sxy * m2y;
    const float c4 = i11 * m2y + 0.5f * sxy * m2x;
    const float c5 = -0.5f * (i00 * m2x * m2x + sxy * m2x * m2y + i11 * m2y * m2y) + logf(norm);

    sval[n][0] = c0; sval[n][1] = c1; sval[n][2] = c2;
    sval[n][3] = c3; sval[n][4] = c4; sval[n][5] = c5;
    sval[n][6] = mult;
    sval[n][7] = colors[n * 3 + 0];
    sval[n][8] = colors[n * 3 + 1];
    sval[n][9] = colors[n * 3 + 2];

    skey[n] = depth;
    sidx[n] = n;
    __syncthreads();

    // --- bitonic sort (ascending by depth), 512 elements, 512 threads ---
    for (int k = 2; k <= NGAUSS; k <<= 1) {
        for (int j = k >> 1; j > 0; j >>= 1) {
            const int ixj = n ^ j;
            if (ixj > n) {
                const bool up = ((n & k) == 0);
                const float ka = skey[n], kbv = skey[ixj];
                if ((ka > kbv) == up) {
                    skey[n] = kbv; skey[ixj] = ka;
                    const int ia = sidx[n]; sidx[n] = sidx[ixj]; sidx[ixj] = ia;
                }
            }
            __syncthreads();
        }
    }

    // --- emit sorted tables ---
    const int src = sidx[n];
    float* coef = ws + WS_COEF;
    coef[n * 8 + 0] = sval[src][0];
    coef[n * 8 + 1] = sval[src][1];
    coef[n * 8 + 2] = sval[src][2];
    coef[n * 8 + 3] = sval[src][3];
    coef[n * 8 + 4] = sval[src][4];
    coef[n * 8 + 5] = sval[src][5];
    coef[n * 8 + 6] = 0.0f;
    coef[n * 8 + 7] = 0.0f;
    ws[WS_MULT + n] = sval[src][6];
    ws[WS_COLS + n * 3 + 0] = sval[src][7];
    ws[WS_COLS + n * 3 + 1] = sval[src][8];
    ws[WS_COLS + n * 3 + 2] = sval[src][9];
}

// ---------------------------------------------------------------------------
// Kernel 2: per-wave rendering of a 16-pixel strip.
// Stage 24KB sorted tables global->LDS once per block (async-LDS path when
// available), then per 16-gaussian tile: 2x V_WMMA_F32_16X16X4_F32 (K=8),
// exp + front-to-back compositing (half-wave scan + shfl_xor merge).
// ---------------------------------------------------------------------------
__global__ __launch_bounds__(256) void gs_render_kernel(
    const float* __restrict__ ws, float* __restrict__ out)
{
    __shared__ float ltab[WS_TOTAL];    // same layout as ws: coef | mult | cols

    // ---- stage tables into LDS (one sweep per block) ----
    const int t = threadIdx.x;
#ifdef GS_USE_ASYNC_LDS
    {
        typedef __attribute__((address_space(1))) i4 gi4;   // prints as "int4 __device__*"
        typedef __attribute__((address_space(3))) i4 li4;   // prints as "int4 __shared__*"
#pragma unroll
        for (int i = 0; i < WS_TOTAL / (4 * 256); ++i) {    // 6 x b128 per thread
            const int e = (t + i * 256) * 4;
            __builtin_amdgcn_global_load_async_to_lds_b128(
                (gi4*)(ws + e), (li4*)(ltab + e), 0, 0);
        }
        __builtin_amdgcn_s_wait_asynccnt(0);
    }
#else
#pragma unroll
    for (int i = 0; i < WS_TOTAL / (4 * 256); ++i) {
        const int e = (t + i * 256) * 4;
        *(v4f*)(ltab + e) = *(const v4f*)(ws + e);
    }
#endif
    __syncthreads();

    const float* lcoef = ltab + WS_COEF;
    const float* lmult = ltab + WS_MULT;
    const float* lcols = ltab + WS_COLS;

    const int lane  = threadIdx.x & 31;
    const int wave  = threadIdx.x >> 5;
    const int strip = blockIdx.x * 8 + wave;        // 16 pixels per wave
    const int p     = strip * 16 + (lane & 15);     // pixel id (duplicated in both halves)

    const float u = (float)(p % IMG_W);             // pixels[...,0] = x = w
    const float v = (float)(p / IMG_W);             // pixels[...,1] = y = h

    // Pixel basis: [u^2, u*v, v^2, u, v, 1, 0, 0]
    float basis[8] = { u * u, u * v, v * v, u, v, 1.0f, 0.0f, 0.0f };

    // 32-bit B fragment (4x16): VGPR0 holds K = (lane<16 ? 0 : 2), VGPR1 K+1.
    const int kb = (lane < 16) ? 0 : 2;
    v2f blo, bhi;
    blo.x = basis[kb];     blo.y = basis[kb + 1];
    bhi.x = basis[kb + 4]; bhi.y = basis[kb + 5];

    const int mrow = lane & 15;                     // A-fragment row (gaussian in tile)
    float T = 1.0f, accR = 0.0f, accG = 0.0f, accB = 0.0f;

    for (int g = 0; g < NGAUSS / 16; ++g) {
        // 32-bit A fragment (16x4): lane holds row mrow, K = kb, kb+1 (+4 for hi slice)
        const int gi = g * 16 + mrow;
        v2f alo, ahi;
        alo.x = lcoef[gi * 8 + kb];     alo.y = lcoef[gi * 8 + kb + 1];
        ahi.x = lcoef[gi * 8 + kb + 4]; ahi.y = lcoef[gi * 8 + kb + 5];

        v8f acc = {};
        acc = __builtin_amdgcn_wmma_f32_16x16x4_f32(false, alo, false, blo,
                                                    (short)0, acc, false, false);
        acc = __builtin_amdgcn_wmma_f32_16x16x4_f32(false, ahi, false, bhi,
                                                    (short)0, acc, false, false);

        // acc[r] = P for gaussian (g*16 + half*8 + r), pixel (lane&15).
        const int mbase = g * 16 + ((lane < 16) ? 0 : 8);
        float Tl = 1.0f, aR = 0.0f, aG = 0.0f, aB = 0.0f;
#pragma unroll
        for (int r = 0; r < 8; ++r) {
            const int idx = mbase + r;
            const float alpha = lmult[idx] * __expf(acc[r]);
            const float w = alpha * Tl;
            aR += w * lcols[idx * 3 + 0];
            aG += w * lcols[idx * 3 + 1];
            aB += w * lcols[idx * 3 + 2];
            Tl *= (1.0f - alpha + 1e-10f);
        }

        // Merge low half (gaussians 0-7) with high half (8-15):
        //   contrib = c_lo + T_lo * c_hi ;  T_tile = T_lo * T_hi
        const float oT = __shfl_xor(Tl, 16, 32);
        const float oR = __shfl_xor(aR, 16, 32);
        const float oG = __shfl_xor(aG, 16, 32);
        const float oB = __shfl_xor(aB, 16, 32);
        const bool hiHalf = (lane >= 16);
        const float loT = hiHalf ? oT : Tl, hiT = hiHalf ? Tl : oT;
        const float loR = hiHalf ? oR : aR, hiR = hiHalf ? aR : oR;
        const float loG = hiHalf ? oG : aG, hiG = hiHalf ? aG : oG;
        const float loB = hiHalf ? oB : aB, hiB = hiHalf ? aB : oB;

        accR += T * (loR + loT * hiR);
        accG += T * (loG + loT * hiG);
        accB += T * (loB + loT * hiB);
        T *= loT * hiT;
    }

    if (lane < 16) {
        out[p * 3 + 0] = accR;
        out[p * 3 + 1] = accG;
        out[p * 3 + 2] = accB;
    }
}

extern "C" void kernel_launch(void* const* d_in, const int* in_sizes, int n_in,
                              void* d_out, int out_size, void* d_ws, size_t ws_size,
                              hipStream_t stream) {
    const float* means3D   = (const float*)d_in[0];
    const float* covs3d    = (const float*)d_in[1];
    const float* colors    = (const float*)d_in[2];
    const float* opacities = (const float*)d_in[3];
    const float* Km        = (const float*)d_in[4];
    const float* Rm        = (const float*)d_in[5];
    const float* tv        = (const float*)d_in[6];
    float* out = (float*)d_out;
    float* ws  = (float*)d_ws;

    gs_prep_sort_kernel<<<1, NGAUSS, 0, stream>>>(
        means3D, covs3d, colors, opacities, Km, Rm, tv, ws);

    // 65536 pixels / 16 per wave = 4096 waves; 8 waves (256 threads) per block.
    gs_render_kernel<<<NPIX / 16 / 8, 256, 0, stream>>>(ws, out);
}